// GraphContinuousPromptModel_36361193128272
// MI455X (gfx1250) — compile-verified
//
#include <hip/hip_runtime.h>

// ---------------------------------------------------------------------------
// 3-layer GIN, dim 128, 100K nodes / 500K edges, MI455X (gfx1250, wave32).
//
// Roofline: edge aggregation moves ~1.5 GB (HBM-bound, ~65us @ 23.3 TB/s) and
// dominates. GEMMs are 39 GFLOP total -> far below the fp32 matrix ceiling,
// so we keep full fp32 precision and run them on the matrix pipe with
// V_WMMA_F32_16X16X4_F32. This revision register-blocks 5 row-tiles per wave
// so every weight (B) fragment feeds 5 WMMAs: 5x fewer weight loads, 5x less
// L2 weight traffic, and 5 independent accumulators to hide XDL latency.
// ---------------------------------------------------------------------------

#define N_NODES 100000
#define N_EDGES 500000
#define IN_DIM  128
#define HID     128
#define HID2    256
#define RT      5               // row-tiles (16 rows each) per block
#define ROWS    (RT * 16)       // 80 rows/block; 100000 = 1250 * 80 exactly
#define LDA     (IN_DIM + 4)    // padded strides: 132 % 64 = 4 banks/row
#define LDH     (HID2  + 4)     // 260 % 64 = 4 -> conflict-free LDS access

typedef float v2f __attribute__((ext_vector_type(2)));
typedef float v8f __attribute__((ext_vector_type(8)));

// ---------------------------------------------------------------------------
// Phase 1: agg[dst] += x[src].  One wave32 per edge, 4 floats per lane.
// unsafeAtomicAdd lowers to global_atomic_add_f32 (no CAS loop).
// ---------------------------------------------------------------------------
__global__ void __launch_bounds__(256) gin_aggregate_kernel(
    const float* __restrict__ x,
    const int*   __restrict__ src,
    const int*   __restrict__ dst,
    float*       __restrict__ agg)
{
  int tid  = blockIdx.x * blockDim.x + threadIdx.x;
  int e    = tid >> 5;
  int lane = tid & 31;
  if (e >= N_EDGES) return;
  int s = src[e];
  int d = dst[e];
  const float4* xs = (const float4*)(x + (size_t)s * IN_DIM);
  float4 v = xs[lane];
  float* ad = agg + (size_t)d * IN_DIM + lane * 4;
  unsafeAtomicAdd(ad + 0, v.x);
  unsafeAtomicAdd(ad + 1, v.y);
  unsafeAtomicAdd(ad + 2, v.z);
  unsafeAtomicAdd(ad + 3, v.w);
}

// ---------------------------------------------------------------------------
// Phase 2: out = ReLU(ReLU((agg + (1+eps)x) @ W1 + b1) @ W2 + b2)
// 8 waves / block, 80-row tile in LDS. Each wave owns a 16-wide N column
// slab and 5 stacked 16x16 accumulators; every B fragment is reused by all 5
// row-tiles (5 WMMAs per 2 weight loads).
//
// f32 16x4 A layout: lanes 0-15 hold M=lane, K={k0,k0+1}; lanes 16-31 hold
// M=lane-16, K={k0+2,k0+3}.  C/D layout: VGPR j -> M=j (lanes 0-15) or
// M=j+8 (lanes 16-31), N = lane & 15.
// ---------------------------------------------------------------------------
__global__ void __launch_bounds__(256) gin_mlp_kernel(
    const float* __restrict__ xin,    // [N,128] current features
    const float* __restrict__ agg,    // [N,128] aggregated neighbors
    const float* __restrict__ W1,     // [128,256] row-major
    const float* __restrict__ b1,     // [256]
    const float* __restrict__ W2,     // [256,128] row-major
    const float* __restrict__ b2,     // [128]
    const float* __restrict__ eps_p,  // scalar (1 elem)
    float*       __restrict__ out)    // [N,128]
{
  __shared__ float sA[ROWS][LDA];
  __shared__ float sH[ROWS][LDH];

  const int tid  = threadIdx.x;
  const int wave = tid >> 5;
  const int lane = tid & 31;
  const int row0 = blockIdx.x * ROWS;
  const float scale = 1.0f + eps_p[0];

  // Stage combined GIN input h0 = agg + (1+eps)*x as float4 streams.
  for (int i = tid; i < ROWS * (IN_DIM / 4); i += 256) {
    int r = i >> 5;            // / 32 float4s per row
    int c = (i & 31) << 2;
    size_t g = (size_t)(row0 + r) * IN_DIM + c;
    float4 va = *(const float4*)(agg + g);
    float4 vx = *(const float4*)(xin + g);
    sA[r][c + 0] = va.x + scale * vx.x;
    sA[r][c + 1] = va.y + scale * vx.y;
    sA[r][c + 2] = va.z + scale * vx.z;
    sA[r][c + 3] = va.w + scale * vx.w;
  }
  __syncthreads();

  const int mrow = lane & 15;          // A-matrix row within a 16-row tile
  const int koff = (lane >> 4) << 1;   // 0 for lanes 0-15, 2 for lanes 16-31
  const int rofs = (lane >> 4) << 3;   // C/D row offset: +8 for upper half

  // ---- GEMM 1: [80x128] @ [128x256] + b1, ReLU -> sH -------------------
  for (int t = 0; t < 2; ++t) {
    const int col = (wave + t * 8) * 16 + mrow;   // N column in [0,256)
    const float bias = b1[col];
    v8f acc[RT];
#pragma unroll
    for (int rt = 0; rt < RT; ++rt)
#pragma unroll
      for (int j = 0; j < 8; ++j) acc[rt][j] = bias;

#pragma unroll 4
    for (int k0 = 0; k0 < IN_DIM; k0 += 4) {
      v2f b;
      b.x = W1[(size_t)(k0 + koff)     * HID2 + col];
      b.y = W1[(size_t)(k0 + koff + 1) * HID2 + col];
#pragma unroll
      for (int rt = 0; rt < RT; ++rt) {
        v2f a = *(const v2f*)&sA[rt * 16 + mrow][k0 + koff]; // ds_load_b64
        acc[rt] = __builtin_amdgcn_wmma_f32_16x16x4_f32(
                      false, a, false, b, (short)0, acc[rt], false, false);
      }
    }
#pragma unroll
    for (int rt = 0; rt < RT; ++rt)
#pragma unroll
      for (int j = 0; j < 8; ++j)
        sH[rt * 16 + j + rofs][col] = fmaxf(acc[rt][j], 0.0f);
  }
  __syncthreads();

  // ---- GEMM 2: [80x256] @ [256x128] + b2, ReLU -> out ------------------
  {
    const int col = wave * 16 + mrow;             // N column in [0,128)
    const float bias = b2[col];
    v8f acc[RT];
#pragma unroll
    for (int rt = 0; rt < RT; ++rt)
#pragma unroll
      for (int j = 0; j < 8; ++j) acc[rt][j] = bias;

#pragma unroll 4
    for (int k0 = 0; k0 < HID2; k0 += 4) {
      v2f b;
      b.x = W2[(size_t)(k0 + koff)     * HID + col];
      b.y = W2[(size_t)(k0 + koff + 1) * HID + col];
#pragma unroll
      for (int rt = 0; rt < RT; ++rt) {
        v2f a = *(const v2f*)&sH[rt * 16 + mrow][k0 + koff]; // ds_load_b64
        acc[rt] = __builtin_amdgcn_wmma_f32_16x16x4_f32(
                      false, a, false, b, (short)0, acc[rt], false, false);
      }
    }
#pragma unroll
    for (int rt = 0; rt < RT; ++rt)
#pragma unroll
      for (int j = 0; j < 8; ++j)
        out[(size_t)(row0 + rt * 16 + j + rofs) * HID + col] =
            fmaxf(acc[rt][j], 0.0f);
  }
}

// ---------------------------------------------------------------------------
extern "C" void kernel_launch(void* const* d_in, const int* in_sizes, int n_in,
                              void* d_out, int out_size, void* d_ws, size_t ws_size,
                              hipStream_t stream) {
  const float* x   = (const float*)d_in[0];
  const int*   ei  = (const int*)  d_in[1];   // [2, N_EDGES]
  const float* W1  = (const float*)d_in[2];   // [3,128,256]
  const float* b1  = (const float*)d_in[3];   // [3,256]
  const float* W2  = (const float*)d_in[4];   // [3,256,128]
  const float* b2  = (const float*)d_in[5];   // [3,128]
  const float* eps = (const float*)d_in[6];   // [3]
  float* out = (float*)d_out;

  const int* src = ei;
  const int* dst = ei + N_EDGES;

  // Workspace: agg plane + one activation plane (2 * 51.2 MB)
  float* agg = (float*)d_ws;
  float* act = agg + (size_t)N_NODES * IN_DIM;

  const size_t featBytes = (size_t)N_NODES * IN_DIM * sizeof(float);
  dim3 aggGrid(((size_t)N_EDGES * 32 + 255) / 256);
  dim3 mlpGrid(N_NODES / ROWS);   // 1250 blocks, exact

  const float* cur = x;
  for (int l = 0; l < 3; ++l) {
    hipMemsetAsync(agg, 0, featBytes, stream);
    gin_aggregate_kernel<<<aggGrid, 256, 0, stream>>>(cur, src, dst, agg);
    // layer 0 -> act, layer 1 -> d_out, layer 2 -> d_out in-place
    // (in-place is safe: each block reads its 80 rows into LDS before
    //  writing those same rows; no cross-block row dependencies)
    float* o = (l == 0) ? act : out;
    gin_mlp_kernel<<<mlpGrid, 256, 0, stream>>>(
        cur, agg,
        W1 + (size_t)l * IN_DIM * HID2,
        b1 + (size_t)l * HID2,
        W2 + (size_t)l * HID2 * HID,
        b2 + (size_t)l * HID,
        eps + l, o);
    cur = o;
  }
}